// DummyDecoder_15083925143701
// MI455X (gfx1250) — compile-verified
//
#include <hip/hip_runtime.h>
#include <hip/hip_bf16.h>

typedef __attribute__((ext_vector_type(2))) float v2f;
typedef __attribute__((ext_vector_type(8))) float v8f;

namespace {
constexpr int   kF      = 17;
constexpr int   kHH     = 300;
constexpr int   kWW     = 400;
constexpr int   kHF     = 38;
constexpr int   kWF     = 50;
constexpr int   kPlane  = kHF * kWF;     // 1900
constexpr int   kNP     = kF * kPlane;   // 32300 seed points
constexpr int   kR      = 5;             // window half-size -> 11x11 active
constexpr float kStride = 8.0f;
constexpr float kVTh    = 0.1f;
constexpr float kInvNb  = 1.0f / 16.0f;  // FACTOR / NEIGHBORS
constexpr float kMaxV   = 1.0f;
}  // namespace

// One wave32 per seed point. Builds the 16x16 splat patch as a rank-3
// outer product with V_WMMA_F32_16X16X4_F32, then scatter-adds the
// 11x11 active region (disc mask + image bounds) with f32 atomics.
__global__ void cif_splat_wmma(const float* __restrict__ cif,
                               float* __restrict__ out) {
  const int wave = blockIdx.x * (blockDim.x >> 5) + (threadIdx.x >> 5);
  if (wave >= kNP) return;                       // wave-uniform guard
  const int lane = threadIdx.x & 31;

  const int f   = wave / kPlane;
  const int rem = wave - f * kPlane;             // h*WF + w
  const float* p = cif + (size_t)f * 5 * kPlane + rem;

  const float v = p[0];                          // broadcast load (uniform)
  if (v <= kVTh) return;                         // wave-uniform: val==0 -> no-op adds

  const float x     = p[1 * kPlane] * kStride;   // indexes WW
  const float y     = p[2 * kPlane] * kStride;   // indexes HH
  const float scale = p[4 * kPlane];

  const float sigma   = fmaxf(1.0f, 0.5f * scale * kStride);
  const float sigma2  = sigma * sigma;
  const float t2s2    = sigma2;                  // TRUNCATE == 1
  const float nh_is2  = -0.5f / sigma2;
  const float val     = v * kInvNb;

  const int cx = (int)floorf(x);
  const int cy = (int)floorf(y);

  const int  idx = lane & 15;        // m (row / dy) for A, n (col / dx) for B
  const bool hi  = lane >= 16;       // high lanes carry K = {2,3}
  const bool act = idx <= 2 * kR;    // 11 active rows/cols, rest zero-padded

  // y-direction factors (A operand, row m = idx)
  const float fy  = (float)(cy - kR + idx) - y;
  const float dy2 = fy * fy;
  const float gy  = act ? expf(dy2 * nh_is2) : 0.0f;
  const float ny  = (act && dy2 < 0.25f) ? 1.0f : 0.0f;

  // x-direction factors (B operand, col n = idx)
  const float fx  = (float)(cx - kR + idx) - x;
  const float dx2 = fx * fx;
  const float gx  = act ? expf(dx2 * nh_is2) : 0.0f;
  const float nx  = (act && dx2 < 0.25f) ? 1.0f : 0.0f;

  // Rank-3 decomposition: val*(gy@gx + ny@nx - (ny*gy)@(nx*gx)); K=3 unused.
  // A 16x4 f32: low lanes VGPR{0,1} = K{0,1}; high lanes = K{2,3}.
  // B  4x16 f32: same K split across lane halves, N = lane&15.
  v2f A, B;
  if (!hi) {
    A[0] = val * gy;        A[1] = val * ny;
    B[0] = gx;              B[1] = nx;
  } else {
    A[0] = -val * ny * gy;  A[1] = 0.0f;
    B[0] = nx * gx;         B[1] = 0.0f;
  }

  v8f acc = {};
  acc = __builtin_amdgcn_wmma_f32_16x16x4_f32(
      /*neg_a=*/false, A, /*neg_b=*/false, B,
      /*c_mod=*/(short)0, acc, /*reuse_a=*/false, /*reuse_b=*/false);

  // C/D layout: lane owns column n = lane&15; VGPR r holds row m = r + (hi?8:0).
  const int xcol = cx - kR + idx;
  if (act && xcol >= 0 && xcol < kWW) {
    float* outf = out + (size_t)f * kHH * kWW + xcol;
    const int mBase = hi ? 8 : 0;
#pragma unroll
    for (int r = 0; r < 8; ++r) {
      const int m = mBase + r;
      if (m <= 2 * kR) {
        const int   yrow = cy - kR + m;
        const float fyy  = (float)yrow - y;
        const float d2   = dx2 + fyy * fyy;        // non-separable disc mask
        if (d2 <= t2s2 && yrow >= 0 && yrow < kHH) {
          atomicAdd(outf + (size_t)yrow * kWW, acc[r]);
        }
      }
    }
  }
}

__global__ void zero_out4(float4* __restrict__ out, int n4) {
  const int i = blockIdx.x * blockDim.x + threadIdx.x;
  if (i < n4) out[i] = make_float4(0.0f, 0.0f, 0.0f, 0.0f);
}

__global__ void clamp_out4(float4* __restrict__ out, int n4) {
  const int i = blockIdx.x * blockDim.x + threadIdx.x;
  if (i < n4) {
    float4 t = out[i];
    t.x = fminf(t.x, kMaxV);
    t.y = fminf(t.y, kMaxV);
    t.z = fminf(t.z, kMaxV);
    t.w = fminf(t.w, kMaxV);
    out[i] = t;
  }
}

extern "C" void kernel_launch(void* const* d_in, const int* in_sizes, int n_in,
                              void* d_out, int out_size, void* d_ws, size_t ws_size,
                              hipStream_t stream) {
  const float* cif = (const float*)d_in[0];   // (17, 5, 38, 50) f32
  // d_in[1] = caf_head, unused by the reference output.
  float* out = (float*)d_out;                 // (17, 300, 400) f32

  // 1) zero the accumulator (d_out is poisoned by the harness)
  const int n4 = out_size / 4;                // 2,040,000 / 4, exact
  zero_out4<<<(n4 + 255) / 256, 256, 0, stream>>>((float4*)out, n4);

  // 2) WMMA splat: 8 waves (one point each) per 256-thread block
  constexpr int kWavesPerBlock = 8;
  const int nblk = (kNP + kWavesPerBlock - 1) / kWavesPerBlock;
  cif_splat_wmma<<<nblk, kWavesPerBlock * 32, 0, stream>>>(cif, out);

  // 3) clamp to MAX_V
  clamp_out4<<<(n4 + 255) / 256, 256, 0, stream>>>((float4*)out, n4);
}